// HandcraftedFPFHExtractor_50044958933383
// MI455X (gfx1250) — compile-verified
//
#include <hip/hip_runtime.h>
#include <hip/hip_bf16.h>
#include <math.h>

typedef __attribute__((ext_vector_type(2))) float v2f;
typedef __attribute__((ext_vector_type(8))) float v8f;

#define NB        11
#define NB3       (NB * NB * NB)   // 1331
#define BATCH     8
#define NPTS      1024
#define TPB       256
#define WAVES     (TPB / 32)       // 8 waves per block
#define ROWTILES  (NPTS / 16)      // 64 row stripes per batch
#define EPSV      1e-6f

// ---------------------------------------------------------------------------
// D = A(16x4) * B(4x16) + C, f32 WMMA. K=4 (3 live dims + zero pad).
// ---------------------------------------------------------------------------
__device__ __forceinline__ v8f gram_wmma(v2f a, v2f b) {
  v8f c = {};
  return __builtin_amdgcn_wmma_f32_16x16x4_f32(
      /*neg_a=*/false, a, /*neg_b=*/false, b,
      /*c_mod=*/(short)0, c, /*reuse_a=*/false, /*reuse_b=*/false);
}

// Branchless fragment build from SoA LDS arrays. ISA 16x4 f32 layout:
//   lanes 0-15 : M=lane,    {K0, K1}
//   lanes16-31 : M=lane-16, {K2, K3(pad=0)}
// All three components are loaded uniformly (stride-1 ds_load_b32, no exec
// masking); the hi/lo split is a v_cndmask select.
__device__ __forceinline__ v2f frag3(const float* __restrict__ sx,
                                     const float* __restrict__ sy,
                                     const float* __restrict__ sz,
                                     int p, bool hi) {
  float qx = sx[p], qy = sy[p], qz = sz[p];
  v2f f;
  f.x = hi ? qz : qx;
  f.y = hi ? 0.0f : qy;
  return f;
}

__global__ void __launch_bounds__(TPB)
fpfh_zero_kernel(float* __restrict__ out, int n) {
  int i = blockIdx.x * TPB + threadIdx.x;
  if (i < n) out[i] = 0.0f;
}

__global__ void __launch_bounds__(TPB)
fpfh_scale_kernel(float* __restrict__ out, int n, float s) {
  int i = blockIdx.x * TPB + threadIdx.x;
  if (i < n) out[i] *= s;
}

// ---------------------------------------------------------------------------
// Main kernel: block = (batch, 16-row stripe). Each of the 8 waves walks 8
// column tiles; per tile it computes six 16x16 Gram matrices via
// v_wmma_f32_16x16x4_f32, then finishes the FPFH angles per pair in VALU and
// scatters into a per-block LDS histogram (ds_add_u32).
// ---------------------------------------------------------------------------
__global__ void __launch_bounds__(TPB)
fpfh_hist_kernel(const float* __restrict__ x, float* __restrict__ out) {
  // SoA point data: conflict-free stride-1 LDS access for fragment builds.
  __shared__ float    sPx[NPTS], sPy[NPTS], sPz[NPTS];
  __shared__ float    sOx[NPTS], sOy[NPTS], sOz[NPTS];
  __shared__ float    sRx[NPTS], sRy[NPTS], sRz[NPTS];  // r = pos x ori
  __shared__ float    sPP[NPTS];   // |pos|^2
  __shared__ float    sOP[NPTS];   // ori . pos
  __shared__ float    sE [NPTS];   // |ori|^2
  __shared__ unsigned sHist[NB3];

  const int tid   = threadIdx.x;
  const int batch = blockIdx.x / ROWTILES;
  const int i0    = (blockIdx.x % ROWTILES) * 16;

  // ---- stage batch points + per-point derived scalars into LDS ----
  for (int p = tid; p < NPTS; p += TPB) {
    const float* src = x + ((size_t)batch * NPTS + p) * 6;
    float px = src[0], py = src[1], pz = src[2];
    float ox = src[3], oy = src[4], oz = src[5];
    sPx[p] = px; sPy[p] = py; sPz[p] = pz;
    sOx[p] = ox; sOy[p] = oy; sOz[p] = oz;
    sRx[p] = py * oz - pz * oy;
    sRy[p] = pz * ox - px * oz;
    sRz[p] = px * oy - py * ox;
    sPP[p] = px * px + py * py + pz * pz;
    sOP[p] = ox * px + oy * py + oz * pz;
    sE[p]  = ox * ox + oy * oy + oz * oz;
  }
  for (int h = tid; h < NB3; h += TPB) sHist[h] = 0u;
  __syncthreads();

  const int  wave    = tid >> 5;
  const int  lane    = tid & 31;
  const int  col     = lane & 15;
  const bool hi      = lane >= 16;
  const int  rowbase = hi ? 8 : 0;

  // Row-side fragments + row scalars: fixed for the whole block.
  const int ir = i0 + col;
  v2f aPos = frag3(sPx, sPy, sPz, ir, hi);
  v2f aOri = frag3(sOx, sOy, sOz, ir, hi);
  v2f aR   = frag3(sRx, sRy, sRz, ir, hi);
  float ppI[8], opI[8], eI[8];
#pragma unroll
  for (int r = 0; r < 8; ++r) {
    int i = i0 + rowbase + r;
    ppI[r] = sPP[i]; opI[r] = sOP[i]; eI[r] = sE[i];
  }

  for (int t = 0; t < ROWTILES / WAVES; ++t) {     // 8 column tiles per wave
    const int j0 = (wave + t * WAVES) * 16;
    const int jc = j0 + col;
    v2f bPos = frag3(sPx, sPy, sPz, jc, hi);
    v2f bOri = frag3(sOx, sOy, sOz, jc, hi);
    v2f bR   = frag3(sRx, sRy, sRz, jc, hi);

    v8f Gpp = gram_wmma(aPos, bPos);   // pos_i . pos_j
    v8f Goo = gram_wmma(aOri, bOri);   // ori_i . ori_j
    v8f Gop = gram_wmma(aOri, bPos);   // ori_i . pos_j
    v8f Gpo = gram_wmma(aPos, bOri);   // pos_i . ori_j
    v8f Gor = gram_wmma(aOri, bR);     // ori_i . r_j
    v8f Gro = gram_wmma(aR,   bOri);   // r_i   . ori_j

    const int   j   = j0 + col;        // this lane's column point
    const float ppJ = sPP[j];
    const float opJ = sOP[j];

#pragma unroll
    for (int r = 0; r < 8; ++r) {
      const int i = i0 + rowbase + r;

      const float a    = fmaxf(ppI[r] + ppJ - 2.0f * Gpp[r], 0.0f); // |delta|^2
      const float dist = sqrtf(a);
      const float bb   = Gop[r] - opI[r];         // ni . delta
      const float dd   = opJ - Gpo[r];            // nj . delta
      const float cc   = Goo[r];                  // ni . nj
      const float ee   = eI[r];                   // |ni|^2
      const float v2   = fmaxf(a * ee - bb * bb, 0.0f); // |delta x ni|^2
      const float sv   = sqrtf(v2);
      const float det  = -(Gor[r] + Gro[r]);      // (delta x ni) . nj
      const float isv  = 1.0f / (sv + EPSV);

      const float alpha = det * isv;
      const float phi   = bb / (dist + EPSV);
      const float sw    = sqrtf(ee) * sv * isv;   // |ni x v_hat|
      const float tnum  = (ee * dd - bb * cc) * isv / (sw + EPSV);
      const float theta = atan2f(tnum, cc) * 0.318309886183790672f; // 1/pi

      int ia = min((int)((alpha + 1.0f) * 0.5f * (float)NB), NB - 1);
      int ip = min((int)((phi   + 1.0f) * 0.5f * (float)NB), NB - 1);
      int it = min((int)((theta + 1.0f) * 0.5f * (float)NB), NB - 1);

      if (i != j)                                  // off-diagonal weights only
        atomicAdd(&sHist[(ia * NB + ip) * NB + it], 1u);
    }
  }

  __syncthreads();
  // Flush: integer-valued float adds (< 2^24) => exact & order-independent.
  float* bout = out + (size_t)batch * NB3;
  for (int h = tid; h < NB3; h += TPB) {
    unsigned c = sHist[h];
    if (c) atomicAdd(bout + h, (float)c);
  }
}

extern "C" void kernel_launch(void* const* d_in, const int* in_sizes, int n_in,
                              void* d_out, int out_size, void* d_ws, size_t ws_size,
                              hipStream_t stream) {
  (void)in_sizes; (void)n_in; (void)d_ws; (void)ws_size; (void)out_size;
  const float* x = (const float*)d_in[0];
  float* out = (float*)d_out;

  const int n = BATCH * NB3;
  fpfh_zero_kernel<<<(n + TPB - 1) / TPB, TPB, 0, stream>>>(out, n);
  fpfh_hist_kernel<<<BATCH * ROWTILES, TPB, 0, stream>>>(x, out);
  const float s = 1.0f / ((float)NPTS * (float)(NPTS - 1));
  fpfh_scale_kernel<<<(n + TPB - 1) / TPB, TPB, 0, stream>>>(out, n, s);
}